// RFDETRProposalGenerator_9216999817760
// MI455X (gfx1250) — compile-verified
//
#include <hip/hip_runtime.h>
#include <hip/hip_bf16.h>

typedef __attribute__((ext_vector_type(2))) float v2f;
typedef __attribute__((ext_vector_type(8))) float v8f;

#define LQ 6
#define BATCH 16
#define NQ 2000
#define HD 256
#define NCLS 80
#define POST_NMS 1000
#define NMS_THRESH 0.7f
#define MROWS (BATCH * NQ) /* 32000 */
#define SORT_N 2048

// ---------------------------------------------------------------------------
// WMMA GEMM: C[M,N] = act(A[M,K] @ W[K,N] + bias[N]), f32 end-to-end via
// V_WMMA_F32_16X16X4_F32. One 16x16 output tile per wave32.
// N, K compile-time -> all inner-loop loads use immediate offsets, no
// divergence anywhere in the K loop (EXEC stays all-1s, as WMMA requires).
// Out-of-range columns (N=4 head) are handled by clamping the column index:
// garbage accumulates only in lanes whose column is never stored.
//
// Fragments per ISA 7.12.2 (f32):
//   A 16x4 : lanes 0-15 rows, VGPRs {K0,K1}; lanes 16-31 rows, {K2,K3}
//   B 4x16 : VGPR0 = K0 (lanes 0-15) / K2 (lanes 16-31); VGPR1 = K1 / K3
//   C 16x16: VGPR g -> row g (lanes 0-15) or 8+g (lanes 16-31), col lane&15
// ---------------------------------------------------------------------------
template <int N, int K, bool RELU>
__global__ __launch_bounds__(256)
void wmma_gemm_bias_act(const float* __restrict__ A, const float* __restrict__ W,
                        const float* __restrict__ bias, float* __restrict__ C,
                        int M)
{
    constexpr int nTilesN = (N + 15) / 16;
    const int wave  = (blockIdx.x * blockDim.x + threadIdx.x) >> 5;
    const int lane  = threadIdx.x & 31;
    const int tileN = wave % nTilesN;
    const int tileM = wave / nTilesN;
    if (tileM * 16 >= M) return;            // wave-uniform exit

    const int lc   = lane & 15;             // A row in tile / B,C column in tile
    const int half = lane >> 4;             // K sub-pair selector (0 -> K{0,1}, 1 -> K{2,3})
    const int aRow = tileM * 16 + lc;
    int bCol = tileN * 16 + lc;
    const bool colOk = (bCol < N);
    if (!colOk) bCol = N - 1;               // clamp: stays in-bounds, result discarded

    const float* __restrict__ Ap = A + (size_t)aRow * K + half * 2;
    const float* __restrict__ Wp = W + (size_t)(half * 2) * N + bCol;
    __builtin_prefetch(Ap, 0, 0);           // global_prefetch_b8

    v8f acc = {};
#pragma unroll
    for (int k = 0; k < K; k += 4) {
        v2f a = *(const v2f*)(Ap + k);      // global_load_b64, immediate offset
        v2f b;
        b.x = Wp[(size_t)k * N];            // row k+2*half   (immediate offset)
        b.y = Wp[(size_t)k * N + N];        // row k+2*half+1 (immediate offset)
        acc = __builtin_amdgcn_wmma_f32_16x16x4_f32(
            /*neg_a=*/false, a, /*neg_b=*/false, b,
            /*c_mod=*/(short)0, acc, /*reuse_a=*/false, /*reuse_b=*/false);
    }

    if (colOk) {
        const float bv = bias[bCol];
        float* __restrict__ Cp = C + (size_t)(tileM * 16 + half * 8) * N + bCol;
#pragma unroll
        for (int g = 0; g < 8; ++g) {
            float v = acc[g] + bv;
            if (RELU) v = fmaxf(v, 0.0f);
            Cp[(size_t)g * N] = v;
        }
    }
}

// ---------------------------------------------------------------------------
// Box reparam decode + objectness/validity -> proposal scores
// ---------------------------------------------------------------------------
__global__ __launch_bounds__(256)
void decode_kernel(const float* __restrict__ delta, const float* __restrict__ refL,
                   const float* __restrict__ logits,
                   const int* __restrict__ img_h, const int* __restrict__ img_w,
                   float* __restrict__ pred_boxes, float* __restrict__ boxes_px,
                   float* __restrict__ scores)
{
    const int i = blockIdx.x * blockDim.x + threadIdx.x;
    if (i >= MROWS) return;

    const float d0 = delta[i * 4 + 0], d1 = delta[i * 4 + 1];
    const float d2 = delta[i * 4 + 2], d3 = delta[i * 4 + 3];
    const float rcx = refL[i * 4 + 0], rcy = refL[i * 4 + 1];
    const float rw  = refL[i * 4 + 2], rh  = refL[i * 4 + 3];

    const float cx = d0 * rw + rcx;
    const float cy = d1 * rh + rcy;
    const float w  = expf(d2) * rw;
    const float h  = expf(d3) * rh;

    pred_boxes[i * 4 + 0] = cx;
    pred_boxes[i * 4 + 1] = cy;
    pred_boxes[i * 4 + 2] = w;
    pred_boxes[i * 4 + 3] = h;

    const float iw = (float)img_w[0], ih = (float)img_h[0];
    const float x1 = (cx - 0.5f * w) * iw;
    const float y1 = (cy - 0.5f * h) * ih;
    const float x2 = (cx + 0.5f * w) * iw;
    const float y2 = (cy + 0.5f * h) * ih;

    boxes_px[i * 4 + 0] = x1;
    boxes_px[i * 4 + 1] = y1;
    boxes_px[i * 4 + 2] = x2;
    boxes_px[i * 4 + 3] = y2;

    const float* lrow = logits + (size_t)i * NCLS;
    float m = lrow[0];
#pragma unroll 4
    for (int c = 1; c < NCLS; ++c) m = fmaxf(m, lrow[c]);

    const bool valid = ((x2 - x1) > 0.0f) && ((y2 - y1) > 0.0f);
    scores[i] = valid ? m : -__builtin_inff();
}

// ---------------------------------------------------------------------------
// Per-batch NMS: bitonic sort (desc) in LDS, sequential greedy suppression,
// blocked scan compaction, write proposals (pad: 0 boxes / -inf score / 0 mask)
// ---------------------------------------------------------------------------
__global__ __launch_bounds__(256)
void nms_kernel(const float* __restrict__ boxes_px, const float* __restrict__ scores,
                float* __restrict__ prop_boxes, float* __restrict__ prop_scores,
                float* __restrict__ prop_mask)
{
    __shared__ float s_score[SORT_N];
    __shared__ int   s_idx[SORT_N];
    __shared__ float s_x1[SORT_N], s_y1[SORT_N], s_x2[SORT_N], s_y2[SORT_N];
    __shared__ unsigned char s_keep[SORT_N];
    __shared__ int   s_rank[SORT_N];
    __shared__ int   s_csum[256];

    const int b   = blockIdx.x;
    const int tid = threadIdx.x;
    const float* bscore = scores + (size_t)b * NQ;
    const float* bbox   = boxes_px + (size_t)b * NQ * 4;

    for (int i = tid; i < SORT_N; i += 256) {
        s_score[i] = (i < NQ) ? bscore[i] : -__builtin_inff();
        s_idx[i]   = i;
    }
    __syncthreads();

    // Bitonic sort, descending by score
    for (unsigned k = 2; k <= SORT_N; k <<= 1) {
        for (unsigned j = k >> 1; j > 0; j >>= 1) {
            for (unsigned i = tid; i < SORT_N; i += 256) {
                const unsigned ixj = i ^ j;
                if (ixj > i) {
                    const bool desc = ((i & k) == 0);
                    const float si = s_score[i], sj = s_score[ixj];
                    const bool doSwap = desc ? (si < sj) : (si > sj);
                    if (doSwap) {
                        s_score[i] = sj; s_score[ixj] = si;
                        const int t = s_idx[i]; s_idx[i] = s_idx[ixj]; s_idx[ixj] = t;
                    }
                }
            }
            __syncthreads();
        }
    }

    // Gather boxes in sorted order; init keep = valid
    for (int i = tid; i < SORT_N; i += 256) {
        const bool inr = (i < NQ);
        const int src  = inr ? s_idx[i] : 0;
        s_x1[i] = inr ? bbox[src * 4 + 0] : 0.0f;
        s_y1[i] = inr ? bbox[src * 4 + 1] : 0.0f;
        s_x2[i] = inr ? bbox[src * 4 + 2] : 0.0f;
        s_y2[i] = inr ? bbox[src * 4 + 3] : 0.0f;
        s_keep[i] = (inr && (s_score[i] > -1e37f)) ? 1 : 0;
    }
    __syncthreads();

    // Greedy sequential suppression; inner j loop parallel across threads
    for (int i = 0; i < NQ; ++i) {
        if (s_keep[i]) {
            const float ax1 = s_x1[i], ay1 = s_y1[i], ax2 = s_x2[i], ay2 = s_y2[i];
            const float aarea = (ax2 - ax1) * (ay2 - ay1);
            for (int j = i + 1 + tid; j < NQ; j += 256) {
                if (s_keep[j]) {
                    const float ix1 = fmaxf(ax1, s_x1[j]);
                    const float iy1 = fmaxf(ay1, s_y1[j]);
                    const float ix2 = fminf(ax2, s_x2[j]);
                    const float iy2 = fminf(ay2, s_y2[j]);
                    const float iw  = fmaxf(ix2 - ix1, 0.0f);
                    const float ihh = fmaxf(iy2 - iy1, 0.0f);
                    const float inter = iw * ihh;
                    const float barea = (s_x2[j] - s_x1[j]) * (s_y2[j] - s_y1[j]);
                    const float uni = aarea + barea - inter;
                    const float iou = (uni > 0.0f) ? (inter / uni) : 0.0f;
                    if (iou > NMS_THRESH) s_keep[j] = 0;
                }
            }
        }
        __syncthreads();
    }

    // Blocked exclusive scan of keep -> rank
    {
        const int base = tid * (SORT_N / 256);
        int c = 0;
#pragma unroll
        for (int u = 0; u < SORT_N / 256; ++u) {
            s_rank[base + u] = c;
            c += s_keep[base + u];
        }
        s_csum[tid] = c;
    }
    __syncthreads();
    if (tid == 0) {
        int acc = 0;
        for (int t = 0; t < 256; ++t) { const int v = s_csum[t]; s_csum[t] = acc; acc += v; }
    }
    __syncthreads();
    {
        const int off  = s_csum[tid];
        const int base = tid * (SORT_N / 256);
#pragma unroll
        for (int u = 0; u < SORT_N / 256; ++u) s_rank[base + u] += off;
    }
    __syncthreads();

    // Write outputs: defaults first, then scatter kept proposals
    float* ob = prop_boxes  + (size_t)b * POST_NMS * 4;
    float* os = prop_scores + (size_t)b * POST_NMS;
    float* om = prop_mask   + (size_t)b * POST_NMS;
    for (int j = tid; j < POST_NMS; j += 256) {
        ob[j * 4 + 0] = 0.0f; ob[j * 4 + 1] = 0.0f;
        ob[j * 4 + 2] = 0.0f; ob[j * 4 + 3] = 0.0f;
        os[j] = -__builtin_inff();
        om[j] = 0.0f;
    }
    __syncthreads();
    for (int j = tid; j < NQ; j += 256) {
        if (s_keep[j]) {
            const int slot = s_rank[j];
            if (slot < POST_NMS) {
                ob[slot * 4 + 0] = s_x1[j];
                ob[slot * 4 + 1] = s_y1[j];
                ob[slot * 4 + 2] = s_x2[j];
                ob[slot * 4 + 3] = s_y2[j];
                os[slot] = s_score[j];
                om[slot] = 1.0f;
            }
        }
    }
}

// ---------------------------------------------------------------------------
extern "C" void kernel_launch(void* const* d_in, const int* in_sizes, int n_in,
                              void* d_out, int out_size, void* d_ws, size_t ws_size,
                              hipStream_t stream)
{
    const float* hs  = (const float*)d_in[0];
    const float* ref = (const float*)d_in[1];
    const float* w1  = (const float*)d_in[2];
    const float* b1  = (const float*)d_in[3];
    const float* w2  = (const float*)d_in[4];
    const float* b2  = (const float*)d_in[5];
    const float* w3  = (const float*)d_in[6];
    const float* b3  = (const float*)d_in[7];
    const float* cw  = (const float*)d_in[8];
    const float* cb  = (const float*)d_in[9];
    const int* img_h = (const int*)d_in[10];
    const int* img_w = (const int*)d_in[11];

    float* out = (float*)d_out;
    float* pred_boxes  = out;                                       // [16,2000,4]
    float* pred_logits = out + (size_t)BATCH * NQ * 4;              // [16,2000,80]
    float* prop_boxes  = pred_logits + (size_t)BATCH * NQ * NCLS;   // [16,1000,4]
    float* prop_scores = prop_boxes + (size_t)BATCH * POST_NMS * 4; // [16,1000]
    float* prop_mask   = prop_scores + (size_t)BATCH * POST_NMS;    // [16,1000]

    float* y1       = (float*)d_ws;                 // [32000,256]
    float* y2       = y1 + (size_t)MROWS * HD;      // [32000,256]
    float* delta    = y2 + (size_t)MROWS * HD;      // [32000,4]
    float* boxes_px = delta + (size_t)MROWS * 4;    // [32000,4]
    float* scoresp  = boxes_px + (size_t)MROWS * 4; // [32000]

    const float* hsL  = hs  + (size_t)(LQ - 1) * MROWS * HD;  // hs[-1]
    const float* refL = ref + (size_t)(LQ - 1) * MROWS * 4;   // ref[-1]

    const int tilesM = MROWS / 16;  // 2000

    // Layer 1: Y1 = relu(hsL @ W1 + b1)   N=256
    {
        const long waves = (long)tilesM * (HD / 16);
        wmma_gemm_bias_act<HD, HD, true><<<(int)((waves + 7) / 8), 256, 0, stream>>>(
            hsL, w1, b1, y1, MROWS);
    }
    // Layer 2: Y2 = relu(Y1 @ W2 + b2)    N=256
    {
        const long waves = (long)tilesM * (HD / 16);
        wmma_gemm_bias_act<HD, HD, true><<<(int)((waves + 7) / 8), 256, 0, stream>>>(
            y1, w2, b2, y2, MROWS);
    }
    // Box head: delta = Y2 @ W3 + b3      N=4 (partial tile, column-clamped)
    {
        const long waves = (long)tilesM * 1;
        wmma_gemm_bias_act<4, HD, false><<<(int)((waves + 7) / 8), 256, 0, stream>>>(
            y2, w3, b3, delta, MROWS);
    }
    // Class head: pred_logits = hsL @ Wc + bc   N=80
    {
        const long waves = (long)tilesM * (NCLS / 16);
        wmma_gemm_bias_act<NCLS, HD, false><<<(int)((waves + 7) / 8), 256, 0, stream>>>(
            hsL, cw, cb, pred_logits, MROWS);
    }

    decode_kernel<<<(MROWS + 255) / 256, 256, 0, stream>>>(
        delta, refL, pred_logits, img_h, img_w, pred_boxes, boxes_px, scoresp);

    nms_kernel<<<BATCH, 256, 0, stream>>>(
        boxes_px, scoresp, prop_boxes, prop_scores, prop_mask);
}